// GraphSage_20538533609946
// MI455X (gfx1250) — compile-verified
//
#include <hip/hip_runtime.h>
#include <math.h>

typedef __attribute__((ext_vector_type(2))) float v2f;
typedef __attribute__((ext_vector_type(8))) float v8f;

#define NFEAT 128
#define NHID  128
#define NCLASS 16

#define G1_ROWS   48     // nodes per block in gemm1 (3 row tiles of 16)
#define G1_TILES  3
#define G1_STRIDE 132    // LDS row stride in floats (pad to avoid bank conflicts)

// ---------------------------------------------------------------- utilities
__global__ void zero_kernel(float* __restrict__ p, long long n) {
  long long i = (long long)blockIdx.x * blockDim.x + threadIdx.x;
  long long stride = (long long)gridDim.x * blockDim.x;
  for (; i < n; i += stride) p[i] = 0.0f;
}

// one wave (32 lanes) per edge; each lane scatters 4 consecutive features
__global__ void scatter_kernel(const float* __restrict__ feat,
                               const int* __restrict__ src,
                               const int* __restrict__ dst,
                               float* __restrict__ agg,
                               float* __restrict__ deg,
                               int addDeg, long long nEdges) {
  long long gid = (long long)blockIdx.x * blockDim.x + threadIdx.x;
  long long e = gid >> 5;
  if (e >= nEdges) return;
  int lane = (int)(gid & 31);
  int s = src[e];
  int d = dst[e];
  int f = lane * 4;
  const float4 v = *(const float4*)(feat + (long long)s * NFEAT + f);
  float* o = agg + (long long)d * NFEAT + f;
  atomicAdd(o + 0, v.x);
  atomicAdd(o + 1, v.y);
  atomicAdd(o + 2, v.z);
  atomicAdd(o + 3, v.w);
  if (addDeg && lane == 0) atomicAdd(deg + d, 1.0f);
}

__global__ void normalize_kernel(float* __restrict__ agg,
                                 const float* __restrict__ deg, long long n) {
  long long i = (long long)blockIdx.x * blockDim.x + threadIdx.x;
  if (i >= n) return;
  float dv = deg[i >> 7];                 // NFEAT == 128
  agg[i] = agg[i] / fmaxf(dv, 1.0f);
}

// ---------------------------------------------------------------- GEMM 1
// h = relu(x @ Wself + agg @ Wneigh + b), all [*,128]x[128,128]
// block: 256 threads = 8 waves; block covers 48 nodes; wave w owns cols [16w,16w+16)
__global__ __launch_bounds__(256)
void gemm1_kernel(const float* __restrict__ x, const float* __restrict__ agg,
                  const float* __restrict__ wself, const float* __restrict__ wneigh,
                  const float* __restrict__ bias, float* __restrict__ h, int nNodes) {
  __shared__ float xs[G1_ROWS * G1_STRIDE];
  __shared__ float as[G1_ROWS * G1_STRIDE];

  const int tid  = threadIdx.x;
  const int lane = tid & 31;
  const int wave = tid >> 5;
  const int half = lane >> 4;     // 0: K+0..1 / rows 0..7 ; 1: K+2..3 / rows 8..15
  const int l16  = lane & 15;
  const int col  = wave * 16 + l16;
  const long long base = (long long)blockIdx.x * G1_ROWS;

  // stage 48x128 tiles of x and agg (float4, guarded with zeros at the tail)
  for (int i = tid; i < G1_ROWS * 32; i += 256) {
    int row = i >> 5;
    int c4  = (i & 31) * 4;
    long long node = base + row;
    float4 vx = make_float4(0.f, 0.f, 0.f, 0.f);
    float4 va = vx;
    if (node < nNodes) {
      vx = *(const float4*)(x   + node * NFEAT + c4);
      va = *(const float4*)(agg + node * NFEAT + c4);
    }
    *(float4*)(&xs[row * G1_STRIDE + c4]) = vx;
    *(float4*)(&as[row * G1_STRIDE + c4]) = va;
  }
  __syncthreads();

  const v8f vzero = {0.f, 0.f, 0.f, 0.f, 0.f, 0.f, 0.f, 0.f};
  v8f acc[G1_TILES];
#pragma unroll
  for (int r = 0; r < G1_TILES; ++r) acc[r] = vzero;

  for (int k0 = 0; k0 < NFEAT; k0 += 4) {
    const int kb = k0 + half * 2;
    // B fragments: B[kb][col], B[kb+1][col]
    v2f bs, bn;
    bs.x = wself [(long long)kb       * NHID + col];
    bs.y = wself [(long long)(kb + 1) * NHID + col];
    bn.x = wneigh[(long long)kb       * NHID + col];
    bn.y = wneigh[(long long)(kb + 1) * NHID + col];
#pragma unroll
    for (int r = 0; r < G1_TILES; ++r) {
      const v2f ax = *(const v2f*)(&xs[(r * 16 + l16) * G1_STRIDE + kb]);
      const v2f aa = *(const v2f*)(&as[(r * 16 + l16) * G1_STRIDE + kb]);
      acc[r] = __builtin_amdgcn_wmma_f32_16x16x4_f32(false, ax, false, bs,
                                                     (short)0, acc[r], false, false);
      acc[r] = __builtin_amdgcn_wmma_f32_16x16x4_f32(false, aa, false, bn,
                                                     (short)0, acc[r], false, false);
    }
  }

  const float bv = bias[col];
#pragma unroll
  for (int r = 0; r < G1_TILES; ++r) {
#pragma unroll
    for (int j = 0; j < 8; ++j) {
      long long node = base + r * 16 + j + half * 8;
      if (node < nNodes) {
        float v = acc[r][j] + bv;
        h[node * NHID + col] = fmaxf(v, 0.0f);
      }
    }
  }
}

// ---------------------------------------------------------------- GEMM 2
// out = log_softmax(h @ Wself2 + agg @ Wneigh2 + b2), [*,128]x[128,16]
// one wave per 16-node tile; one 16x16 C tile covers all 16 classes
__global__ __launch_bounds__(256)
void gemm2_kernel(const float* __restrict__ h, const float* __restrict__ agg,
                  const float* __restrict__ wself, const float* __restrict__ wneigh,
                  const float* __restrict__ bias, float* __restrict__ out, int nNodes) {
  const int lane = threadIdx.x & 31;
  const int wave = threadIdx.x >> 5;
  const int half = lane >> 4;
  const int l16  = lane & 15;
  const long long base = ((long long)blockIdx.x * 8 + wave) * 16;
  if (base >= nNodes) return;          // wave-uniform exit: EXEC stays all-ones

  long long nodeA = base + l16;
  if (nodeA >= nNodes) nodeA = nNodes - 1;   // clamp; garbage rows discarded at store

  v8f acc = {0.f, 0.f, 0.f, 0.f, 0.f, 0.f, 0.f, 0.f};

  for (int k0 = 0; k0 < NHID; k0 += 4) {
    const int kb = k0 + half * 2;
    const v2f ah = *(const v2f*)(h   + nodeA * NHID + kb);
    const v2f aa = *(const v2f*)(agg + nodeA * NHID + kb);
    v2f bs, bn;
    bs.x = wself [kb       * NCLASS + l16];
    bs.y = wself [(kb + 1) * NCLASS + l16];
    bn.x = wneigh[kb       * NCLASS + l16];
    bn.y = wneigh[(kb + 1) * NCLASS + l16];
    acc = __builtin_amdgcn_wmma_f32_16x16x4_f32(false, ah, false, bs,
                                                (short)0, acc, false, false);
    acc = __builtin_amdgcn_wmma_f32_16x16x4_f32(false, aa, false, bn,
                                                (short)0, acc, false, false);
  }

  const float bv = bias[l16];
#pragma unroll
  for (int j = 0; j < 8; ++j) {
    float v = acc[j] + bv;               // row = j + half*8, col = l16
    float mx = v;
#pragma unroll
    for (int off = 8; off >= 1; off >>= 1)
      mx = fmaxf(mx, __shfl_xor(mx, off, 16));
    float e = expf(v - mx);
    float s = e;
#pragma unroll
    for (int off = 8; off >= 1; off >>= 1)
      s += __shfl_xor(s, off, 16);
    long long node = base + j + half * 8;
    if (node < nNodes) out[node * NCLASS + l16] = v - mx - logf(s);
  }
}

// ---------------------------------------------------------------- launcher
extern "C" void kernel_launch(void* const* d_in, const int* in_sizes, int n_in,
                              void* d_out, int out_size, void* d_ws, size_t ws_size,
                              hipStream_t stream) {
  const float* x   = (const float*)d_in[0];
  const int*   src = (const int*)  d_in[1];
  const int*   dst = (const int*)  d_in[2];
  const float* ws1 = (const float*)d_in[3];
  const float* wn1 = (const float*)d_in[4];
  const float* b1  = (const float*)d_in[5];
  const float* ws2 = (const float*)d_in[6];
  const float* wn2 = (const float*)d_in[7];
  const float* b2  = (const float*)d_in[8];
  float* out = (float*)d_out;

  const int nNodes = in_sizes[0] / NFEAT;
  const long long nEdges = in_sizes[1];

  char* w = (char*)d_ws;
  size_t degBytes = (((size_t)nNodes * 4) + 255) & ~(size_t)255;
  float* deg = (float*)w;
  float* agg = (float*)(w + degBytes);
  float* h   = agg + (size_t)nNodes * NFEAT;

  const long long aggN  = (long long)nNodes * NFEAT;
  const long long zeroN = (long long)(degBytes / 4) + aggN;

  const int scatterBlocks = (int)((nEdges * 32 + 255) / 256);
  const int normBlocks    = (int)((aggN + 255) / 256);

  // ---- layer 1 ----
  zero_kernel<<<2048, 256, 0, stream>>>((float*)w, zeroN);
  scatter_kernel<<<scatterBlocks, 256, 0, stream>>>(x, src, dst, agg, deg, 1, nEdges);
  normalize_kernel<<<normBlocks, 256, 0, stream>>>(agg, deg, aggN);
  gemm1_kernel<<<(nNodes + G1_ROWS - 1) / G1_ROWS, 256, 0, stream>>>(
      x, agg, ws1, wn1, b1, h, nNodes);

  // ---- layer 2 ----
  zero_kernel<<<2048, 256, 0, stream>>>(agg, aggN);
  scatter_kernel<<<scatterBlocks, 256, 0, stream>>>(h, src, dst, agg, deg, 0, nEdges);
  normalize_kernel<<<normBlocks, 256, 0, stream>>>(agg, deg, aggN);
  gemm2_kernel<<<(nNodes + 127) / 128, 256, 0, stream>>>(
      h, agg, ws2, wn2, b2, out, nNodes);
}